// Shallow_RGCN_88648124990150
// MI455X (gfx1250) — compile-verified
//
#include <hip/hip_runtime.h>
#include <hip/hip_bf16.h>

// Shallow RGCN forward for MI455X (gfx1250, wave32).
// GEMMs on V_WMMA_F32_16X16X4_F32 (exact fp32), edge phase as float4
// gathers + global_atomic_add_f32 scatters (L2-resident working set).

typedef __attribute__((ext_vector_type(2))) float v2f;
typedef __attribute__((ext_vector_type(8))) float v8f;

#define HD 64   // EMBED_DIM == HIDDEN_DIM == 64

// ---------------------------------------------------------------------------
// dst[n, 0:64] = embed[x[n]] @ W (64x64 row-major) (+ bias)
// One wave32 per 16-node tile. A fragments per ISA 16x4 f32 layout:
//   lane = (row, half); A vgpr pair for K-chunk c = h[row][4c+2*half .. +1]
// B fragment: lane&15 = output col within tile, half selects K rows.
// D: vgpr v -> row (v + 8*half), col (lane&15 + 16*nt).
// ---------------------------------------------------------------------------
__global__ __launch_bounds__(256)
void rgcn_gemm_wmma(const int* __restrict__ x,
                    const float* __restrict__ embed,
                    const float* __restrict__ W,
                    const float* __restrict__ bias,
                    float* __restrict__ dst,
                    int N)
{
    const int wave  = (int)((blockIdx.x * blockDim.x + threadIdx.x) >> 5);
    const int lane  = threadIdx.x & 31;
    const int tiles = (N + 15) >> 4;
    if (wave >= tiles) return;

    const int node_base = wave << 4;
    const int row  = lane & 15;
    const int half = lane >> 4;
    const bool full_tile = (node_base + 16 <= N);   // wave-uniform

    // A fragments: 16 K-chunks of the 16x64 activation tile (kept in VGPRs,
    // reused for all 4 output column tiles and shared across the k-loop).
    int anode = node_base + row;
    if (anode >= N) anode = N - 1;                  // clamp for tail tile
    const float* hrow = embed + (size_t)x[anode] * HD;

    v2f A[16];
#pragma unroll
    for (int c = 0; c < 16; ++c) {
        A[c] = *(const v2f*)(hrow + 4 * c + 2 * half);
    }

#pragma unroll
    for (int nt = 0; nt < 4; ++nt) {
        const int col = nt * 16 + row;              // B/D column for this lane
        v8f acc = {};
#pragma unroll
        for (int c = 0; c < 16; ++c) {
            const int k0 = 4 * c + 2 * half;
            v2f b;
            b.x = W[(k0 + 0) * HD + col];
            b.y = W[(k0 + 1) * HD + col];
            acc = __builtin_amdgcn_wmma_f32_16x16x4_f32(
                      /*neg_a=*/false, A[c],
                      /*neg_b=*/false, b,
                      /*c_mod=*/(short)0, acc,
                      /*reuse_a=*/false, /*reuse_b=*/false);
        }
        const float bv = bias ? bias[col] : 0.0f;
        float* dcol = dst + (size_t)(node_base + 8 * half) * HD + col;
        if (full_tile) {
            // Common case: straight-line stores, no per-element exec masking.
#pragma unroll
            for (int v = 0; v < 8; ++v) {
                dcol[(size_t)v * HD] = acc[v] + bv;
            }
        } else {
#pragma unroll
            for (int v = 0; v < 8; ++v) {
                const int m = node_base + v + 8 * half;
                if (m < N) dst[(size_t)m * HD + col] = acc[v] + bv;
            }
        }
    }
}

// ---------------------------------------------------------------------------
// cnt[r][dst] += 1.0 per edge (reference: segment_sum of mask).
// ---------------------------------------------------------------------------
__global__ __launch_bounds__(256)
void rgcn_count(const int* __restrict__ edst,
                const int* __restrict__ etype,
                float* __restrict__ cnt,   // [NUM_REL][N]
                int E, int N)
{
    const int e = blockIdx.x * blockDim.x + threadIdx.x;
    if (e >= E) return;
    atomicAdd(&cnt[(size_t)etype[e] * N + edst[e]], 1.0f);
}

// ---------------------------------------------------------------------------
// out[dst] += hr[src] / max(cnt_r[dst], 1) for edges of relation r.
// 16 lanes per edge, float4 per lane (coalesced b128 gather + 4 f32 atomics).
// ---------------------------------------------------------------------------
__global__ __launch_bounds__(256)
void rgcn_scatter(const int* __restrict__ esrc,
                  const int* __restrict__ edst,
                  const int* __restrict__ etype,
                  int rel,
                  const float* __restrict__ hr,     // [N][64]
                  const float* __restrict__ cnt_r,  // [N] (already offset by r)
                  float* __restrict__ out,          // [N][64]
                  int E)
{
    const long t = (long)blockIdx.x * blockDim.x + threadIdx.x;
    const long e = t >> 4;
    if (e >= E) return;
    if (etype[e] != rel) return;
    const int j = (int)(t & 15);

    const int s = esrc[e];
    const int d = edst[e];
    const float inv = 1.0f / fmaxf(cnt_r[d], 1.0f);

    const float4 v = *(const float4*)(hr + (size_t)s * HD + 4 * j);
    float* o = out + (size_t)d * HD + 4 * j;
    atomicAdd(o + 0, v.x * inv);
    atomicAdd(o + 1, v.y * inv);
    atomicAdd(o + 2, v.z * inv);
    atomicAdd(o + 3, v.w * inv);
}

// ---------------------------------------------------------------------------
// ReLU + global mean pool accumulation: 16 lanes per node, float4 per lane.
// ---------------------------------------------------------------------------
__global__ __launch_bounds__(256)
void rgcn_pool(const float* __restrict__ out,
               const int* __restrict__ batch,
               float* __restrict__ g_sum,   // [G][64]
               float* __restrict__ g_cnt,   // [G]
               int N)
{
    const long t = (long)blockIdx.x * blockDim.x + threadIdx.x;
    const long n = t >> 4;
    if (n >= N) return;
    const int j = (int)(t & 15);

    const int g = batch[n];
    const float4 v = *(const float4*)(out + (size_t)n * HD + 4 * j);
    float* gs = g_sum + (size_t)g * HD + 4 * j;
    atomicAdd(gs + 0, fmaxf(v.x, 0.0f));
    atomicAdd(gs + 1, fmaxf(v.y, 0.0f));
    atomicAdd(gs + 2, fmaxf(v.z, 0.0f));
    atomicAdd(gs + 3, fmaxf(v.w, 0.0f));
    if (j == 0) atomicAdd(&g_cnt[g], 1.0f);
}

// ---------------------------------------------------------------------------
// logits[g] = (g_sum[g]/max(g_cnt[g],1)) @ lin_W (64x2) + lin_b
// ---------------------------------------------------------------------------
__global__ __launch_bounds__(256)
void rgcn_head(const float* __restrict__ g_sum,
               const float* __restrict__ g_cnt,
               const float* __restrict__ lin_W,  // [64][2]
               const float* __restrict__ lin_b,  // [2]
               float* __restrict__ logits,       // [G][2]
               int G)
{
    const int g = blockIdx.x * blockDim.x + threadIdx.x;
    if (g >= G) return;
    const float inv = 1.0f / fmaxf(g_cnt[g], 1.0f);
    float a0 = lin_b[0];
    float a1 = lin_b[1];
#pragma unroll 8
    for (int k = 0; k < HD; ++k) {
        const float e = g_sum[(size_t)g * HD + k] * inv;
        a0 += e * lin_W[2 * k + 0];
        a1 += e * lin_W[2 * k + 1];
    }
    logits[2 * g + 0] = a0;
    logits[2 * g + 1] = a1;
}

// ---------------------------------------------------------------------------
extern "C" void kernel_launch(void* const* d_in, const int* in_sizes, int n_in,
                              void* d_out, int out_size, void* d_ws, size_t ws_size,
                              hipStream_t stream)
{
    const int*   x         = (const int*)  d_in[0];
    const int*   edge_idx  = (const int*)  d_in[1];   // [2][E] flat
    const int*   edge_type = (const int*)  d_in[2];
    const int*   batch     = (const int*)  d_in[3];
    const float* embed     = (const float*)d_in[4];
    const float* W_rel     = (const float*)d_in[5];   // [R][64][64]
    const float* W_root    = (const float*)d_in[6];   // [64][64]
    const float* conv_bias = (const float*)d_in[7];
    const float* lin_W     = (const float*)d_in[8];
    const float* lin_b     = (const float*)d_in[9];

    const int N = in_sizes[0];
    const int E = in_sizes[2];
    const int R = in_sizes[5] / (HD * HD);            // NUM_REL (= 3)
    const int G = out_size / 2;                       // NUM_GRAPHS (= 512)

    const int* esrc = edge_idx;
    const int* edst = edge_idx + E;

    // Workspace layout (~52.5 MB): out | hr | cnt[R][N] | g_sum | g_cnt
    float* outbuf = (float*)d_ws;
    float* hr     = outbuf + (size_t)N * HD;
    float* cnt    = hr     + (size_t)N * HD;
    float* g_sum  = cnt    + (size_t)R * N;
    float* g_cnt  = g_sum  + (size_t)G * HD;

    // Zero the accumulators (cnt .. g_cnt, contiguous).
    hipMemsetAsync(cnt, 0,
                   ((size_t)R * N + (size_t)G * HD + (size_t)G) * sizeof(float),
                   stream);

    const int tiles       = (N + 15) >> 4;            // 16-node tiles
    const int gemm_blocks = (tiles + 7) >> 3;         // 8 waves / 256-thr block

    // out = h @ W_root + conv_bias  (writes every element; no pre-zero needed)
    rgcn_gemm_wmma<<<gemm_blocks, 256, 0, stream>>>(x, embed, W_root, conv_bias,
                                                    outbuf, N);

    // Per-relation in-degree counts (single edge pass for all relations).
    rgcn_count<<<(E + 255) / 256, 256, 0, stream>>>(edst, edge_type, cnt, E, N);

    const long sc_threads = (long)E * 16;
    const int  sc_blocks  = (int)((sc_threads + 255) / 256);

    for (int r = 0; r < R; ++r) {
        // hr = h @ W_rel[r]   (buffer reused across relations)
        rgcn_gemm_wmma<<<gemm_blocks, 256, 0, stream>>>(
            x, embed, W_rel + (size_t)r * HD * HD, nullptr, hr, N);
        // out[dst] += hr[src] / max(cnt_r[dst], 1)
        rgcn_scatter<<<sc_blocks, 256, 0, stream>>>(
            esrc, edst, edge_type, r, hr, cnt + (size_t)r * N, outbuf, E);
    }

    // ReLU + graph mean pooling accumulation.
    const long pl_threads = (long)N * 16;
    rgcn_pool<<<(int)((pl_threads + 255) / 256), 256, 0, stream>>>(
        outbuf, batch, g_sum, g_cnt, N);

    // Head: [G,64] @ [64,2] + b -> d_out
    rgcn_head<<<(G + 255) / 256, 256, 0, stream>>>(
        g_sum, g_cnt, lin_W, lin_b, (float*)d_out, G);
}